// GCNNet_764504179050
// MI455X (gfx1250) — compile-verified
//
#include <hip/hip_runtime.h>
#include <math.h>

typedef __attribute__((ext_vector_type(2))) float v2f;
typedef __attribute__((ext_vector_type(8))) float v8f;

// ---------------------------------------------------------------------------
// Norm precompute kernels
// ---------------------------------------------------------------------------

__global__ void fill_deg_kernel(float* __restrict__ deg, int n) {
  int i = blockIdx.x * blockDim.x + threadIdx.x;
  if (i < n) deg[i] = 1.0f;  // self-loop weight
}

__global__ void deg_accum_kernel(const int* __restrict__ dst,
                                 const float* __restrict__ ew,
                                 float* __restrict__ deg, int E) {
  int e = blockIdx.x * blockDim.x + threadIdx.x;
  if (e < E) atomicAdd(&deg[dst[e]], ew[e]);
}

// deg -> dis (in place) and selfnorm = dis^2
__global__ void dis_selfnorm_kernel(float* __restrict__ deg,
                                    float* __restrict__ selfnorm, int n) {
  int i = blockIdx.x * blockDim.x + threadIdx.x;
  if (i < n) {
    float d = deg[i];
    float r = (d > 0.0f) ? rsqrtf(d) : 0.0f;
    deg[i] = r;
    selfnorm[i] = r * r;
  }
}

__global__ void edge_norm_kernel(const int* __restrict__ src,
                                 const int* __restrict__ dst,
                                 const float* __restrict__ ew,
                                 const float* __restrict__ dis,
                                 float* __restrict__ norm, int E) {
  int e = blockIdx.x * blockDim.x + threadIdx.x;
  if (e < E) norm[e] = dis[src[e]] * ew[e] * dis[dst[e]];
}

// ---------------------------------------------------------------------------
// WMMA f32 GEMM:  H[nrows x NOUT] = X[nrows x K] @ W[K x NOUT]
// One wave computes a 16-row stripe across all column tiles.
//
// W is staged in LDS in a k-pair-interleaved layout so each B fragment
// (W[ka][n], W[ka+1][n]) is one contiguous, 8B-aligned ds_load_b64:
//   Wlds[(kp*NP + n)*2 + j] = W[2*kp + j][n],  kp = k/2, j in {0,1}
// Zero-padded to NP = ceil16(NOUT) columns.
// ---------------------------------------------------------------------------

template <int K, int NOUT>
__global__ __launch_bounds__(256) void gemm_wmma_kernel(
    const float* __restrict__ X, const float* __restrict__ W,
    float* __restrict__ H, int nrows) {
  constexpr int CT = (NOUT + 15) / 16;  // column tiles
  constexpr int NP = CT * 16;           // padded width in LDS
  __shared__ float Wlds[K * NP];        // interpreted as [K/2][NP][2]

  // Cooperative load of W into interleaved LDS layout with zero padding.
  for (int idx = threadIdx.x; idx < K * NP; idx += blockDim.x) {
    const int j = idx & 1;          // which of the k-pair
    const int t = idx >> 1;
    const int n = t % NP;           // column
    const int kp = t / NP;          // k-pair index
    const int r = 2 * kp + j;       // source row of W
    Wlds[idx] = (n < NOUT) ? W[r * NOUT + n] : 0.0f;
  }
  __syncthreads();

  const int lane = threadIdx.x & 31;
  const int wave = threadIdx.x >> 5;
  const int m0 = (blockIdx.x * 8 + wave) * 16;
  if (m0 >= nrows) return;

  const int kk = lane >> 4;   // 0: K+0/K+1, 1: K+2/K+3  (A & B frag halves)
  const int ln = lane & 15;   // M index for A, N index for B/C

  int arow = m0 + ln;
  if (arow >= nrows) arow = nrows - 1;  // partial-tile clamp

  v8f acc[CT] = {};

  const float* __restrict__ xrow = X + (size_t)arow * K;
  // Per-lane base into the interleaved LDS W: k-pair (k/2 + kk), column ln.
  const float* __restrict__ wbase = Wlds + (size_t)2 * (kk * NP + ln);

  for (int k = 0; k < K; k += 4) {
    const int ka = k + 2 * kk;
    // A fragment: lane holds A[arow][ka], A[arow][ka+1]   (8B aligned)
    v2f a = *(const v2f*)(xrow + ka);
    const float* __restrict__ wk = wbase + (size_t)(k / 2) * NP * 2;
#pragma unroll
    for (int t = 0; t < CT; ++t) {
      // B fragment: one contiguous 8B LDS load = {W[ka][t*16+ln], W[ka+1][..]}
      v2f b = *(const v2f*)(wk + t * 32);
      acc[t] = __builtin_amdgcn_wmma_f32_16x16x4_f32(
          /*neg_a=*/false, a, /*neg_b=*/false, b,
          /*c_mod=*/(short)0, acc[t], /*reuse_a=*/false, /*reuse_b=*/false);
    }
  }

  // Store: VGPR v, lane L -> row m0 + v + 8*(L>>4), col t*16 + (L&15)
  if (m0 + 16 <= nrows) {
    // Full tile: unguarded row stores.
#pragma unroll
    for (int t = 0; t < CT; ++t) {
      const int col = t * 16 + ln;
      if (col < NOUT) {
#pragma unroll
        for (int v = 0; v < 8; ++v) {
          const int r = m0 + v + 8 * kk;
          H[(size_t)r * NOUT + col] = acc[t][v];
        }
      }
    }
  } else {
#pragma unroll
    for (int t = 0; t < CT; ++t) {
      const int col = t * 16 + ln;
      if (col < NOUT) {
#pragma unroll
        for (int v = 0; v < 8; ++v) {
          const int r = m0 + v + 8 * kk;
          if (r < nrows) H[(size_t)r * NOUT + col] = acc[t][v];
        }
      }
    }
  }
}

// ---------------------------------------------------------------------------
// Aggregation kernels
// ---------------------------------------------------------------------------

// agg[i][f] = bias[f] + selfnorm[i] * h[i][f]
__global__ void agg_init_kernel(const float* __restrict__ h,
                                const float* __restrict__ selfnorm,
                                const float* __restrict__ bias,
                                float* __restrict__ agg, int n, int F) {
  int idx = blockIdx.x * blockDim.x + threadIdx.x;
  if (idx < n * F) {
    int i = idx / F;
    int f = idx - i * F;
    agg[idx] = bias[f] + selfnorm[i] * h[idx];
  }
}

// One wave per edge; lanes stride over features. Coalesced gather of h[src],
// global_atomic_add_f32 scatter into agg[dst] (L2-resident).
__global__ __launch_bounds__(256) void scatter_kernel(
    const float* __restrict__ h, const int* __restrict__ src,
    const int* __restrict__ dst, const float* __restrict__ norm,
    float* __restrict__ agg, int E, int F) {
  const int wave_id = blockIdx.x * (blockDim.x >> 5) + (threadIdx.x >> 5);
  const int lane = threadIdx.x & 31;
  if (wave_id >= E) return;
  const int s = src[wave_id];
  const int d = dst[wave_id];
  const float w = norm[wave_id];
  const float* __restrict__ hs = h + (size_t)s * F;
  float* __restrict__ ad = agg + (size_t)d * F;
  for (int f = lane; f < F; f += 32) {
    atomicAdd(&ad[f], w * hs[f]);
  }
}

__global__ void relu_kernel(float* __restrict__ x, int total) {
  int idx = blockIdx.x * blockDim.x + threadIdx.x;
  if (idx < total) x[idx] = fmaxf(x[idx], 0.0f);
}

// ---------------------------------------------------------------------------
// log_softmax over C=40 classes, one wave per row (lane f covers f, f+32)
// ---------------------------------------------------------------------------
__global__ __launch_bounds__(256) void log_softmax40_kernel(
    const float* __restrict__ in, float* __restrict__ out, int n) {
  const int row = blockIdx.x * (blockDim.x >> 5) + (threadIdx.x >> 5);
  const int lane = threadIdx.x & 31;
  if (row >= n) return;
  const float* __restrict__ r = in + (size_t)row * 40;
  const float v0 = r[lane];                                     // f = lane < 40
  const float v1 = (lane < 8) ? r[32 + lane] : -__builtin_inff();
  float m = fmaxf(v0, v1);
  for (int o = 16; o > 0; o >>= 1) m = fmaxf(m, __shfl_xor(m, o, 32));
  float s = __expf(v0 - m) + ((lane < 8) ? __expf(v1 - m) : 0.0f);
  for (int o = 16; o > 0; o >>= 1) s += __shfl_xor(s, o, 32);
  const float ls = __logf(s);
  float* __restrict__ o_ = out + (size_t)row * 40;
  o_[lane] = v0 - m - ls;
  if (lane < 8) o_[32 + lane] = v1 - m - ls;
}

// ---------------------------------------------------------------------------
// Launcher
// ---------------------------------------------------------------------------
extern "C" void kernel_launch(void* const* d_in, const int* in_sizes, int n_in,
                              void* d_out, int out_size, void* d_ws,
                              size_t ws_size, hipStream_t stream) {
  const float* x = (const float*)d_in[0];
  const int* edge_index = (const int*)d_in[1];
  const float* ew = (const float*)d_in[2];
  const float* W0 = (const float*)d_in[3];
  const float* b0 = (const float*)d_in[4];
  const float* W1 = (const float*)d_in[5];
  const float* b1 = (const float*)d_in[6];
  const float* W2 = (const float*)d_in[7];
  const float* b2 = (const float*)d_in[8];

  const int E = in_sizes[1] / 2;
  const int H = in_sizes[4];        // 64
  const int Fin = in_sizes[3] / H;  // 256
  const int N = in_sizes[0] / Fin;  // 100000
  (void)n_in; (void)out_size; (void)ws_size;

  const int* src = edge_index;
  const int* dst = edge_index + E;

  // Workspace layout (floats), each segment rounded to 64-float alignment.
  auto align64 = [](size_t v) { return (v + 63) & ~(size_t)63; };
  float* ws = (float*)d_ws;
  size_t off = 0;
  float* dis = ws + off;      off = align64(off + (size_t)N);
  float* selfnorm = ws + off; off = align64(off + (size_t)N);
  float* norm = ws + off;     off = align64(off + (size_t)E);
  float* hbuf = ws + off;     off = align64(off + (size_t)N * 64);
  float* abuf = ws + off;     off = align64(off + (size_t)N * 64);

  const int T256 = 256;
  const int mtiles = (N + 15) / 16;
  const int gemm_blocks = (mtiles + 7) / 8;
  const int edge_wave_blocks = (E + 7) / 8;  // 8 waves per 256-thread block
  const int row_wave_blocks = (N + 7) / 8;

  // ---- edge norm (computed once, shared by all 3 layers) ----
  fill_deg_kernel<<<(N + T256 - 1) / T256, T256, 0, stream>>>(dis, N);
  deg_accum_kernel<<<(E + T256 - 1) / T256, T256, 0, stream>>>(dst, ew, dis, E);
  dis_selfnorm_kernel<<<(N + T256 - 1) / T256, T256, 0, stream>>>(dis, selfnorm, N);
  edge_norm_kernel<<<(E + T256 - 1) / T256, T256, 0, stream>>>(src, dst, ew, dis, norm, E);

  // ---- layer 0: x[N,256] -> abuf[N,64] ----
  gemm_wmma_kernel<256, 64><<<gemm_blocks, T256, 0, stream>>>(x, W0, hbuf, N);
  agg_init_kernel<<<((size_t)N * 64 + T256 - 1) / T256, T256, 0, stream>>>(
      hbuf, selfnorm, b0, abuf, N, 64);
  scatter_kernel<<<edge_wave_blocks, T256, 0, stream>>>(hbuf, src, dst, norm, abuf, E, 64);
  relu_kernel<<<((size_t)N * 64 + T256 - 1) / T256, T256, 0, stream>>>(abuf, N * 64);

  // ---- layer 1: abuf[N,64] -> abuf[N,64] (hbuf as intermediate) ----
  gemm_wmma_kernel<64, 64><<<gemm_blocks, T256, 0, stream>>>(abuf, W1, hbuf, N);
  agg_init_kernel<<<((size_t)N * 64 + T256 - 1) / T256, T256, 0, stream>>>(
      hbuf, selfnorm, b1, abuf, N, 64);
  scatter_kernel<<<edge_wave_blocks, T256, 0, stream>>>(hbuf, src, dst, norm, abuf, E, 64);
  relu_kernel<<<((size_t)N * 64 + T256 - 1) / T256, T256, 0, stream>>>(abuf, N * 64);

  // ---- layer 2: abuf[N,64] -> abuf[N,40] (compact stride 40) ----
  gemm_wmma_kernel<64, 40><<<gemm_blocks, T256, 0, stream>>>(abuf, W2, hbuf, N);
  agg_init_kernel<<<((size_t)N * 40 + T256 - 1) / T256, T256, 0, stream>>>(
      hbuf, selfnorm, b2, abuf, N, 40);
  scatter_kernel<<<edge_wave_blocks, T256, 0, stream>>>(hbuf, src, dst, norm, abuf, E, 40);

  // ---- log_softmax into d_out ----
  log_softmax40_kernel<<<row_wave_blocks, T256, 0, stream>>>(abuf, (float*)d_out, N);
}